// AverageHausdorffLoss_26018911879197
// MI455X (gfx1250) — compile-verified
//
#include <hip/hip_runtime.h>
#include <math.h>

typedef float v2f __attribute__((ext_vector_type(2)));
typedef float v8f __attribute__((ext_vector_type(8)));

#define KPTS 4096
#define NT   256      // 4096 / 16 tiles
#define HW   65536    // 256*256
#define NIMG 16
#define SENT 1.0e9f
#define GRP  4        // g-tiles per wave per pass (4 WMMAs share one B frag)

// ---------------------------------------------------------------------------
// Kernel 1: binarize, 3x3-erosion edge detect, wave32 ordered compaction.
// One wave (32 lanes) per (image, source) pair; raster order preserved via
// ballot prefix-ranks so truncation-at-4096 matches jnp.nonzero(size=4096).
// ---------------------------------------------------------------------------
__global__ __launch_bounds__(32)
void edge_compact_kernel(const float* __restrict__ gth,
                         const float* __restrict__ pred,
                         float2* __restrict__ gpts,
                         float2* __restrict__ ppts,
                         int* __restrict__ counts) {
  const int blk = blockIdx.x;           // 0..31
  const int img = blk & 15;
  const int src = blk >> 4;             // 0 = gth, 1 = pred
  const float* base = (src == 0 ? gth : pred) + (size_t)img * HW;
  float2* pts = (src == 0 ? gpts : ppts) + (size_t)img * KPTS;
  const int lane = threadIdx.x;

  // sentinel-fill: padded slots are far away, never win a min
  for (int i = lane; i < KPTS; i += 32) pts[i] = make_float2(SENT, SENT);

  int count = 0;
  for (int it = 0; it < HW / 32; ++it) {
    const int p = it * 32 + lane;
    const int y = p >> 8;
    const int x = p & 255;
    bool edge = false;
    if (base[p] > 0.5f) {
      // min-pool pads with +inf, so OOB neighbors never create edges:
      // edge <=> center==1 and some in-bounds 3x3 neighbor ==0
      for (int dy = -1; dy <= 1 && !edge; ++dy) {
        const int yy = y + dy;
        if (yy < 0 || yy > 255) continue;
        for (int dx = -1; dx <= 1; ++dx) {
          const int xx = x + dx;
          if (xx < 0 || xx > 255) continue;
          if (!(base[yy * 256 + xx] > 0.5f)) { edge = true; break; }
        }
      }
    }
    const unsigned mask = (unsigned)__ballot(edge);   // wave32
    if (edge) {
      const int pos = count + __popc(mask & ((1u << lane) - 1u));
      if (pos < KPTS) pts[pos] = make_float2((float)y, (float)x);
    }
    count += __popc(mask);
  }
  if (lane == 0) counts[blk] = count;   // full count (may exceed KPTS)
}

// ---------------------------------------------------------------------------
// Kernel 2: per-plane symmetric Hausdorff via V_WMMA_F32_16X16X4_F32.
// dist^2 tile (16 gth rows x 16 pred cols) from a single rank-4 WMMA:
//   A row m  = ( gy, gx, gy^2+gx^2, 1 )
//   B col n  = (-2py, -2px, 1, py^2+px^2 )
// All intermediates < 2^24 for real coords -> exact integers, matching JAX.
//
// All min-reductions run on the float BIT PATTERNS as signed ints: every
// consumed dist^2 is a non-negative finite f32, whose numeric order equals
// the signed-int order of its bits. v_min_i32/v_min3_i32 need none of the
// sNaN-quieting v_max(x,x) canonicalizations that fminf lowering inserts.
//
// 4-way g-tile blocking: 4 WMMAs share each B fragment, their column
// partial-mins combine in registers -> one ds_min_i32 per B tile, issued
// branch-free by all 32 lanes (EXEC stays all-1s, as WMMA requires).
// ---------------------------------------------------------------------------
__global__ __launch_bounds__(256)
void hausdorff_wmma_kernel(const float2* __restrict__ gpts,
                           const float2* __restrict__ ppts,
                           const int* __restrict__ counts,
                           float* __restrict__ losses) {
  __shared__ float2 s_p[KPTS];        // 32 KB: pred points (hot, inner loop)
  __shared__ int    s_dpmin2[KPTS];   // 16 KB: float bits of per-col min d^2
  __shared__ float  s_dgsum, s_dpsum;

  const int img  = blockIdx.x;
  const int tid  = threadIdx.x;       // 256 threads = 8 waves
  const int lane = tid & 31;
  const int wave = tid >> 5;
  const int INITBITS = __float_as_int(3.0e38f);   // positive bits: int-max init

  for (int i = tid; i < KPTS; i += 256) {
    s_p[i]      = ppts[(size_t)img * KPTS + i];
    s_dpmin2[i] = INITBITS;
  }
  if (tid == 0) { s_dgsum = 0.0f; s_dpsum = 0.0f; }
  __syncthreads();

  const int nG  = counts[img];
  const int nP  = counts[NIMG + img];
  const int nGv = nG < KPTS ? nG : KPTS;   // valid slots (ref truncates at 4096)
  const int nPv = nP < KPTS ? nP : KPTS;

  const bool hi  = lane >= 16;
  const int  l16 = lane & 15;
  const float2* gimg = gpts + (size_t)img * KPTS;

  float dgLocal = 0.0f;

  // 256 g-tiles = 8 waves * 8 passes * GRP(4) tiles
  for (int tg = 0; tg < NT / (8 * GRP); ++tg) {
    const int t0 = (tg * 8 + wave) * GRP;   // first of 4 consecutive g tiles

    v2f a[GRP];
    int rowmin[GRP][8];                     // float bits, int-min reduced
#pragma unroll
    for (int g = 0; g < GRP; ++g) {
      const float2 gp = gimg[(t0 + g) * 16 + l16];
      // A layout (16x4 f32): lanes 0-15 hold K={0,1}, lanes 16-31 K={2,3}
      a[g][0] = hi ? (gp.x * gp.x + gp.y * gp.y) : gp.x;
      a[g][1] = hi ? 1.0f                        : gp.y;
#pragma unroll
      for (int r = 0; r < 8; ++r) rowmin[g][r] = INITBITS;
    }

    const float2* pit = s_p + l16;
    for (int u = 0; u < NT; ++u, pit += 16) {   // all p col tiles
      const float2 pp = *pit;
      // B layout (4x16 f32): lanes 0-15 hold K={0,1}, lanes 16-31 K={2,3}
      v2f b;
      b[0] = hi ? 1.0f                        : (-2.0f * pp.x);
      b[1] = hi ? (pp.x * pp.x + pp.y * pp.y) : (-2.0f * pp.y);

      int cm = INITBITS;
#pragma unroll
      for (int g = 0; g < GRP; ++g) {
        v8f c = {0.f, 0.f, 0.f, 0.f, 0.f, 0.f, 0.f, 0.f};
        v8f d = __builtin_amdgcn_wmma_f32_16x16x4_f32(
            false, a[g], false, b, (short)0, c, false, false);
#pragma unroll
        for (int r = 0; r < 8; ++r) {
          const int di = __float_as_int(d[r]);
          cm = min(cm, di);                            // column partial min
          rowmin[g][r] = min(rowmin[g][r], di);        // row min accumulation
        }
      }
      // combine lane halves: every lane then holds the full column min
      cm = min(cm, __shfl_xor(cm, 16, 32));
      // branch-free: lanes l and l+16 write the same value to the same slot
      atomicMin(&s_dpmin2[u * 16 + l16], cm);
    }

    // cross-lane min within each 16-lane half -> per-row min over all p
#pragma unroll
    for (int g = 0; g < GRP; ++g) {
#pragma unroll
      for (int m = 1; m <= 8; m <<= 1) {
#pragma unroll
        for (int r = 0; r < 8; ++r)
          rowmin[g][r] = min(rowmin[g][r], __shfl_xor(rowmin[g][r], m, 32));
      }
    }
    if (lane == 0 || lane == 16) {
#pragma unroll
      for (int g = 0; g < GRP; ++g) {
        const int rbase = (t0 + g) * 16 + (hi ? 8 : 0);
#pragma unroll
        for (int r = 0; r < 8; ++r) {
          const int row = rbase + r;
          if (row < nGv)
            dgLocal += sqrtf(__int_as_float(rowmin[g][r]));  // sqrt(min d^2)
        }
      }
    }
  }
  if (lane == 0 || lane == 16) atomicAdd(&s_dgsum, dgLocal);
  __syncthreads();

  float dpLocal = 0.0f;
  for (int n = tid; n < KPTS; n += 256)
    if (n < nPv) dpLocal += sqrtf(__int_as_float(s_dpmin2[n]));
  atomicAdd(&s_dpsum, dpLocal);
  __syncthreads();

  if (tid == 0) {
    float loss;
    if (nG == 0 || nP == 0) {
      // reference yields NaN whenever either point set is empty
      loss = __builtin_nanf("");
    } else {
      const float g2p = s_dgsum / (float)nG;   // divide by FULL counts
      const float p2g = s_dpsum / (float)nP;
      const float ahd = 0.5f * (g2p + p2g);
      loss = 1.0f - 1.0f / (1.0f + ahd);
    }
    losses[img] = loss;
  }
}

// ---------------------------------------------------------------------------
// Kernel 3: nanmean over the 16 per-plane losses.
// ---------------------------------------------------------------------------
__global__ __launch_bounds__(32)
void finalize_kernel(const float* __restrict__ losses, float* __restrict__ out) {
  const int lane = threadIdx.x;
  float v = 0.0f, c = 0.0f;
  if (lane < NIMG) {
    const float x = losses[lane];
    if (!(x != x)) { v = x; c = 1.0f; }
  }
#pragma unroll
  for (int m = 1; m <= 16; m <<= 1) {
    v += __shfl_xor(v, m, 32);
    c += __shfl_xor(c, m, 32);
  }
  if (lane == 0) out[0] = (c > 0.0f) ? (v / c) : __builtin_nanf("");
}

// ---------------------------------------------------------------------------
// Workspace layout (needs ~1 MB + 192 B):
//   [0, 512K)        gth edge points, 16 x 4096 x float2 (y,x)
//   [512K, 1M)       pred edge points
//   [1M, 1M+128)     counts[32]  (g: 0..15, p: 16..31)
//   [1M+128, +64)    per-plane losses[16]
// ---------------------------------------------------------------------------
extern "C" void kernel_launch(void* const* d_in, const int* in_sizes, int n_in,
                              void* d_out, int out_size, void* d_ws, size_t ws_size,
                              hipStream_t stream) {
  const float* gth  = (const float*)d_in[0];
  const float* pred = (const float*)d_in[1];
  char* ws = (char*)d_ws;
  float2* gpts  = (float2*)ws;
  float2* ppts  = (float2*)(ws + (size_t)524288);
  int*    counts = (int*)(ws + (size_t)1048576);
  float*  losses = (float*)(ws + (size_t)1048576 + 128);

  edge_compact_kernel<<<32, 32, 0, stream>>>(gth, pred, gpts, ppts, counts);
  hausdorff_wmma_kernel<<<NIMG, 256, 0, stream>>>(gpts, ppts, counts, losses);
  finalize_kernel<<<1, 32, 0, stream>>>(losses, (float*)d_out);
}